// CGNNExpert_20538533609917
// MI455X (gfx1250) — compile-verified
//
#include <hip/hip_runtime.h>
#include <hip/hip_bf16.h>
#include <math.h>

// GATv2 GNN (N=10000, E=320000, HID=256, HEADS=16, C=16, L=3), fp32.
//  - GEMMs on the CDNA5 matrix pipe via V_WMMA_F32_16X16X4_F32 (fp32-exact),
//    16x64 strip per wave (4 accumulators reusing one A fragment).
//  - Static dst-CSR built once (degree/scan/fill/sort -> deterministic order);
//    softmax + aggregation + residual fused into ONE gather kernel per layer:
//    one wave32 per (dst, head). No floating-point atomics anywhere.
// Workspace need: ~64 MB.

#define HIDN   256
#define NHEADS 16
#define CHAN   16
#define NLAY   3
#define INDIM  128

typedef float v2f __attribute__((ext_vector_type(2)));
typedef float v8f __attribute__((ext_vector_type(8)));

__device__ __forceinline__ float gelu_exact(float x) {
  return 0.5f * x * (1.0f + erff(x * 0.70710678118654752f));
}

// ---------------------------------------------------------------------------
// Tiled GEMM: C[M,256] = act(A[M,K] @ W[K,256] + bias).
// block = 128 (4 waves); wave w owns the 16x64 strip at cols w*64.
// Each K-step loads one A fragment (v2f) and feeds 4 WMMAs.
// ---------------------------------------------------------------------------
template <int ACT>
__global__ __launch_bounds__(128) void gemm16_wmma(
    const float* __restrict__ A, const float* __restrict__ W,
    const float* __restrict__ bias, float* __restrict__ C,
    int M, int K) {
  const int wave = threadIdx.x >> 5;
  const int lane = threadIdx.x & 31;
  const int half = lane >> 4;  // 0: K pair {0,1}, 1: K pair {2,3}
  const int l16  = lane & 15;
  const int row0 = blockIdx.x * 16;
  const int col0 = wave * 64;

  int ar = row0 + l16;
  if (ar >= M) ar = M - 1;  // clamp loads; stores guarded below
  const float* __restrict__ arow = A + (size_t)ar * K;
  const float* __restrict__ wp   = W + col0 + l16;

  v8f acc[4] = {};
#pragma unroll 2
  for (int k = 0; k < K; k += 4) {
    const int ka = k + half * 2;
    // A 16x4: lane&15 = M row; VGPR0 holds K=ka, VGPR1 K=ka+1
    v2f a = *(const v2f*)(arow + ka);
    const float* p0 = wp + (size_t)ka * HIDN;
    const float* p1 = p0 + HIDN;
    v2f b[4];
#pragma unroll
    for (int j = 0; j < 4; ++j) {
      b[j].x = p0[16 * j];
      b[j].y = p1[16 * j];
    }
#pragma unroll
    for (int j = 0; j < 4; ++j)
      acc[j] = __builtin_amdgcn_wmma_f32_16x16x4_f32(
          false, a, false, b[j], (short)0, acc[j], false, false);
  }
  // C layout: acc[j][i] -> row = row0 + i + half*8, col = col0 + 16j + l16
#pragma unroll
  for (int j = 0; j < 4; ++j) {
    const int c  = col0 + 16 * j + l16;
    const float bs = bias[c];
#pragma unroll
    for (int i = 0; i < 8; ++i) {
      const int r = row0 + i + half * 8;
      if (r < M) {
        float v = acc[j][i] + bs;
        if (ACT == 1) v = gelu_exact(v);
        C[(size_t)r * HIDN + c] = v;
      }
    }
  }
}

// ---------------------------------------------------------------------------
// LayerNorm over last dim (256). One wave32 per row, 8 elems per lane.
// ---------------------------------------------------------------------------
__global__ __launch_bounds__(256) void ln_kernel(
    const float* __restrict__ X, const float* __restrict__ g,
    const float* __restrict__ b, float* __restrict__ Y, int M) {
  const int row  = blockIdx.x * (blockDim.x >> 5) + (threadIdx.x >> 5);
  const int lane = threadIdx.x & 31;
  if (row >= M) return;
  const float* __restrict__ x = X + (size_t)row * HIDN;
  float v[8], s = 0.f, s2 = 0.f;
#pragma unroll
  for (int i = 0; i < 8; ++i) {
    v[i] = x[lane + i * 32];
    s  += v[i];
    s2 += v[i] * v[i];
  }
#pragma unroll
  for (int off = 16; off; off >>= 1) {
    s  += __shfl_xor(s,  off, 32);
    s2 += __shfl_xor(s2, off, 32);
  }
  const float mu  = s * (1.f / HIDN);
  const float var = s2 * (1.f / HIDN) - mu * mu;
  const float inv = rsqrtf(var + 1e-5f);
  float* __restrict__ y = Y + (size_t)row * HIDN;
#pragma unroll
  for (int i = 0; i < 8; ++i) {
    const int c = lane + i * 32;
    y[c] = (v[i] - mu) * inv * g[c] + b[c];
  }
}

// ---------------------------------------------------------------------------
// CSR build (edge list is static across layers).
// ---------------------------------------------------------------------------
__global__ __launch_bounds__(256) void degree_kernel(
    const int* __restrict__ dst, int* __restrict__ counts, int E, int EN) {
  const int e = blockIdx.x * blockDim.x + threadIdx.x;
  if (e >= EN) return;
  atomicAdd(&counts[(e < E) ? dst[e] : (e - E)], 1);
}

// single-block exclusive scan: off[0..n] from counts[0..n-1]
__global__ __launch_bounds__(256) void scan_offsets_kernel(
    const int* __restrict__ counts, int* __restrict__ off, int n) {
  __shared__ int part[256];
  const int t = threadIdx.x;
  const int chunk = (n + 255) / 256;
  const int b0 = t * chunk;
  int b1 = b0 + chunk; if (b1 > n) b1 = n;
  int s = 0;
  for (int i = b0; i < b1; ++i) s += counts[i];
  part[t] = s;
  __syncthreads();
  for (int o = 1; o < 256; o <<= 1) {  // Hillis-Steele inclusive scan
    const int v = (t >= o) ? part[t - o] : 0;
    __syncthreads();
    part[t] += v;
    __syncthreads();
  }
  int excl = (t == 0) ? 0 : part[t - 1];
  for (int i = b0; i < b1; ++i) { off[i] = excl; excl += counts[i]; }
  if (t == 255) off[n] = part[255];
}

__global__ __launch_bounds__(256) void csr_fill_kernel(
    const int* __restrict__ dst, int* __restrict__ cursor,
    int* __restrict__ eid, int E, int EN) {
  const int e = blockIdx.x * blockDim.x + threadIdx.x;
  if (e >= EN) return;
  const int d = (e < E) ? dst[e] : (e - E);
  eid[atomicAdd(&cursor[d], 1)] = e;
}

// insertion-sort each bucket -> deterministic summation order (deg ~ 33)
__global__ __launch_bounds__(256) void sort_buckets_kernel(
    const int* __restrict__ off, int* __restrict__ eid, int Nn) {
  const int d = blockIdx.x * blockDim.x + threadIdx.x;
  if (d >= Nn) return;
  const int beg = off[d], end = off[d + 1];
  for (int i = beg + 1; i < end; ++i) {
    const int key = eid[i];
    int j = i - 1;
    while (j >= beg && eid[j] > key) { eid[j + 1] = eid[j]; --j; }
    eid[j + 1] = key;
  }
}

// ---------------------------------------------------------------------------
// Edge pass: logits[e,h] = sum_c leaky_relu(xl[src]+xr[dst]+ew*We)*att.
// One thread per (edge, head); self loops for e in [E, E+N).
// ---------------------------------------------------------------------------
__global__ __launch_bounds__(256) void edge_logits_kernel(
    const int* __restrict__ src, const int* __restrict__ dst,
    const float* __restrict__ ew, const float* __restrict__ ewm,
    const float* __restrict__ xl, const float* __restrict__ xr,
    const float* __restrict__ We_l, const float* __restrict__ att_l,
    float* __restrict__ logits, int E, int T) {
  const int t = blockIdx.x * blockDim.x + threadIdx.x;
  if (t >= T) return;
  const int e = t >> 4, h = t & 15;
  int s, d; float w;
  if (e < E) { s = src[e]; d = dst[e]; w = ew[e]; }
  else       { s = e - E;  d = s;      w = ewm[0]; }
  const float4* __restrict__ pl = (const float4*)(xl + (size_t)s * HIDN + h * CHAN);
  const float4* __restrict__ pr = (const float4*)(xr + (size_t)d * HIDN + h * CHAN);
  const float4* __restrict__ pe = (const float4*)(We_l + h * CHAN);
  const float4* __restrict__ pa = (const float4*)(att_l + h * CHAN);
  float logit = 0.f;
#pragma unroll
  for (int q = 0; q < 4; ++q) {
    const float4 a = pl[q], r = pr[q], eb = pe[q], at = pa[q];
    float u;
    u = a.x + r.x + w * eb.x; u = (u > 0.f) ? u : 0.2f * u; logit += u * at.x;
    u = a.y + r.y + w * eb.y; u = (u > 0.f) ? u : 0.2f * u; logit += u * at.y;
    u = a.z + r.z + w * eb.z; u = (u > 0.f) ? u : 0.2f * u; logit += u * at.z;
    u = a.w + r.w + w * eb.w; u = (u > 0.f) ? u : 0.2f * u; logit += u * at.w;
  }
  logits[t] = logit;
}

// ---------------------------------------------------------------------------
// Fused segment softmax + aggregation + residual. One wave32 per (dst, head):
//   mx = max_e logit ; sm = sum_e exp(logit-mx)
//   h[d,h,:] += sum_e xl[src,h,:]*exp(logit-mx)/sm + out_b[h,:]
// Lanes 0-15 = channels (even edge slots), lanes 16-31 = channels (odd slots).
// ---------------------------------------------------------------------------
__global__ __launch_bounds__(256) void gat_aggregate_kernel(
    const int* __restrict__ src, const int* __restrict__ off,
    const int* __restrict__ eid, const float* __restrict__ logits,
    const float* __restrict__ xl, const float* __restrict__ ob,
    float* __restrict__ H, int E, int Nn) {
  const int wid = blockIdx.x * (blockDim.x >> 5) + (threadIdx.x >> 5);
  if (wid >= Nn * NHEADS) return;
  const int d = wid >> 4, h = wid & 15;
  const int lane = threadIdx.x & 31;
  const int beg = off[d], end = off[d + 1];

  float mx = -3.402823466e38f;
  for (int i = beg + lane; i < end; i += 32)
    mx = fmaxf(mx, logits[(size_t)eid[i] * NHEADS + h]);
#pragma unroll
  for (int o = 16; o; o >>= 1) mx = fmaxf(mx, __shfl_xor(mx, o, 32));

  float sm = 0.f;
  for (int i = beg + lane; i < end; i += 32)
    sm += expf(logits[(size_t)eid[i] * NHEADS + h] - mx);
#pragma unroll
  for (int o = 16; o; o >>= 1) sm += __shfl_xor(sm, o, 32);

  const int c = lane & 15, par = lane >> 4;
  float acc = 0.f;
  for (int i = beg + par; i < end; i += 2) {
    const int e = eid[i];
    const int s = (e < E) ? src[e] : (e - E);
    const float a = expf(logits[(size_t)e * NHEADS + h] - mx);
    acc += xl[(size_t)s * HIDN + h * CHAN + c] * a;
  }
  acc += __shfl_xor(acc, 16, 32);
  if (lane < 16)
    H[(size_t)d * HIDN + h * CHAN + c] += acc / sm + ob[h * CHAN + c];
}

// mean edge weight, single block -> deterministic
__global__ __launch_bounds__(256) void ewmean_kernel(
    const float* __restrict__ ew, float* __restrict__ ewm, int E) {
  __shared__ float sb[256];
  float s = 0.f;
  for (int i = threadIdx.x; i < E; i += 256) s += ew[i];
  sb[threadIdx.x] = s;
  __syncthreads();
  for (int o = 128; o; o >>= 1) {
    if (threadIdx.x < (unsigned)o) sb[threadIdx.x] += sb[threadIdx.x + o];
    __syncthreads();
  }
  if (threadIdx.x == 0) ewm[0] = sb[0] / (float)E;
}

// per-block partial column sums of h (coalesced)
__global__ __launch_bounds__(256) void colsum_partial_kernel(
    const float* __restrict__ H, float* __restrict__ part, int M) {
  const int c = threadIdx.x;
  const int chunk = (M + gridDim.x - 1) / gridDim.x;
  const int r0 = blockIdx.x * chunk;
  int r1 = r0 + chunk; if (r1 > M) r1 = M;
  float s = 0.f;
  for (int r = r0; r < r1; ++r) s += H[(size_t)r * HIDN + c];
  part[(size_t)blockIdx.x * HIDN + c] = s;
}

// ctxm = (sum of partials)/M ; gate = sigmoid(ctxm @ Wg + bg)
__global__ __launch_bounds__(256) void gate_kernel(
    const float* __restrict__ part, const float* __restrict__ Wg,
    const float* __restrict__ bg, float* __restrict__ ctxm,
    float* __restrict__ gate, int M, int npart) {
  __shared__ float ctx[HIDN];
  const int c = threadIdx.x;
  float s = 0.f;
  for (int b = 0; b < npart; ++b) s += part[(size_t)b * HIDN + c];
  const float cm = s / (float)M;
  ctx[c] = cm;
  ctxm[c] = cm;
  __syncthreads();
  float g = bg[c];
  for (int k = 0; k < HIDN; ++k) g += ctx[k] * Wg[(size_t)k * HIDN + c];
  gate[c] = 1.f / (1.f + expf(-g));
}

// y = LN(h + gate*ctxm). One wave32 per row.
__global__ __launch_bounds__(256) void final_ln_kernel(
    const float* __restrict__ H, const float* __restrict__ gate,
    const float* __restrict__ ctxm, const float* __restrict__ g,
    const float* __restrict__ b, float* __restrict__ Y, int M) {
  const int row  = blockIdx.x * (blockDim.x >> 5) + (threadIdx.x >> 5);
  const int lane = threadIdx.x & 31;
  if (row >= M) return;
  const float* __restrict__ x = H + (size_t)row * HIDN;
  float v[8], s = 0.f, s2 = 0.f;
#pragma unroll
  for (int i = 0; i < 8; ++i) {
    const int c = lane + i * 32;
    v[i] = x[c] + gate[c] * ctxm[c];
    s  += v[i];
    s2 += v[i] * v[i];
  }
#pragma unroll
  for (int off = 16; off; off >>= 1) {
    s  += __shfl_xor(s,  off, 32);
    s2 += __shfl_xor(s2, off, 32);
  }
  const float mu  = s * (1.f / HIDN);
  const float var = s2 * (1.f / HIDN) - mu * mu;
  const float inv = rsqrtf(var + 1e-5f);
  float* __restrict__ y = Y + (size_t)row * HIDN;
#pragma unroll
  for (int i = 0; i < 8; ++i) {
    const int c = lane + i * 32;
    y[c] = (v[i] - mu) * inv * g[c] + b[c];
  }
}

// ---------------------------------------------------------------------------
extern "C" void kernel_launch(void* const* d_in, const int* in_sizes, int n_in,
                              void* d_out, int out_size, void* d_ws,
                              size_t ws_size, hipStream_t stream) {
  const float* x     = (const float*)d_in[0];
  const int*   eidx  = (const int*)d_in[1];
  const float* ew    = (const float*)d_in[2];
  const float* W_in  = (const float*)d_in[3];
  const float* b_in  = (const float*)d_in[4];
  const float* g_in  = (const float*)d_in[5];
  const float* bn_in = (const float*)d_in[6];
  const float* g_ly  = (const float*)d_in[7];
  const float* bn_ly = (const float*)d_in[8];
  const float* Wl    = (const float*)d_in[9];
  const float* bl    = (const float*)d_in[10];
  const float* Wr    = (const float*)d_in[11];
  const float* br    = (const float*)d_in[12];
  const float* We    = (const float*)d_in[13];
  const float* att   = (const float*)d_in[14];
  const float* out_b = (const float*)d_in[15];
  const float* Wg    = (const float*)d_in[16];
  const float* bg    = (const float*)d_in[17];
  const float* g_fn  = (const float*)d_in[18];
  const float* b_fn  = (const float*)d_in[19];

  const int E  = in_sizes[2];          // edge_weight is [E,1]
  const int Nn = in_sizes[0] / INDIM;  // nodes
  const int EN = E + Nn;               // edges incl. self loops
  const int* src = eidx;
  const int* dst = eidx + E;
  const int NPART = 64;

  // workspace carve-up
  float* ws = (float*)d_ws;
  const size_t NH = (size_t)Nn * HIDN;
  float* h       = ws;
  float* hn      = h + NH;
  float* xl      = hn + NH;
  float* xr      = xl + NH;
  float* logits  = xr + NH;                         // EN*NHEADS
  float* colpart = logits + (size_t)EN * NHEADS;    // NPART*HIDN
  float* ctxm    = colpart + (size_t)NPART * HIDN;  // HIDN
  float* gatep   = ctxm + HIDN;                     // HIDN
  float* ewm     = gatep + HIDN;                    // 1
  int*   counts  = (int*)(ewm + 1);                 // Nn
  int*   off     = counts + Nn;                     // Nn+1
  int*   cursor  = off + Nn + 1;                    // Nn
  int*   eid     = cursor + Nn;                     // EN

  const int egrid_e = (EN + 255) / 256;

  // ---- build dst-CSR once (deterministic after per-bucket sort) ----
  hipMemsetAsync(counts, 0, (size_t)Nn * sizeof(int), stream);
  degree_kernel<<<egrid_e, 256, 0, stream>>>(dst, counts, E, EN);
  scan_offsets_kernel<<<1, 256, 0, stream>>>(counts, off, Nn);
  hipMemcpyAsync(cursor, off, (size_t)Nn * sizeof(int),
                 hipMemcpyDeviceToDevice, stream);
  csr_fill_kernel<<<egrid_e, 256, 0, stream>>>(dst, cursor, eid, E, EN);
  sort_buckets_kernel<<<(Nn + 255) / 256, 256, 0, stream>>>(off, eid, Nn);
  ewmean_kernel<<<1, 256, 0, stream>>>(ew, ewm, E);

  // ---- input projection: h = LN(gelu(x @ W_in + b_in)) ----
  gemm16_wmma<1><<<(Nn + 15) / 16, 128, 0, stream>>>(x, W_in, b_in, hn, Nn,
                                                     INDIM);
  ln_kernel<<<(Nn + 7) / 8, 256, 0, stream>>>(hn, g_in, bn_in, h, Nn);

  const int T = EN * NHEADS;
  const int egrid_t = (T + 255) / 256;
  const int agrid = (Nn * NHEADS + 7) / 8;  // 8 waves per block

  for (int l = 0; l < NLAY; ++l) {
    ln_kernel<<<(Nn + 7) / 8, 256, 0, stream>>>(h, g_ly + l * HIDN,
                                                bn_ly + l * HIDN, hn, Nn);
    gemm16_wmma<0><<<(Nn + 15) / 16, 128, 0, stream>>>(
        hn, Wl + (size_t)l * HIDN * HIDN, bl + l * HIDN, xl, Nn, HIDN);
    gemm16_wmma<0><<<(Nn + 15) / 16, 128, 0, stream>>>(
        hn, Wr + (size_t)l * HIDN * HIDN, br + l * HIDN, xr, Nn, HIDN);

    edge_logits_kernel<<<egrid_t, 256, 0, stream>>>(
        src, dst, ew, ewm, xl, xr, We + (size_t)l * HIDN,
        att + (size_t)l * NHEADS * CHAN, logits, E, T);
    gat_aggregate_kernel<<<agrid, 256, 0, stream>>>(
        src, off, eid, logits, xl, out_b + l * HIDN, h, E, Nn);
  }

  // ---- global context gating + final LN -> d_out ----
  colsum_partial_kernel<<<NPART, 256, 0, stream>>>(h, colpart, Nn);
  gate_kernel<<<1, 256, 0, stream>>>(colpart, Wg, bg, ctxm, gatep, Nn, NPART);
  final_ln_kernel<<<(Nn + 7) / 8, 256, 0, stream>>>(h, gatep, ctxm, g_fn, b_fn,
                                                    (float*)d_out, Nn);
}